// SIPostProcess_59828894433886
// MI455X (gfx1250) — compile-verified
//
#include <hip/hip_runtime.h>
#include <stdint.h>

// Volume: N=8, C=1, D=48, H=256, W=256 (NCDHW, fp32)
#define W_   256
#define H_   256
#define D_   48
#define N_   8
#define TOT_ (N_ * D_ * H_ * W_)   // 25165824

// Tile: 64 (x) * 8 (y) * 8 (z) outputs per 256-thread block
#define TX_  64
#define TY_  8
#define TZ_  8
// Halo tile (+1 each side)
#define HX_  (TX_ + 2)             // 66
#define HY_  (TY_ + 2)             // 10
#define HZ_  (TZ_ + 2)             // 10
#define HN_  (HX_ * HY_ * HZ_)     // 6600 floats = 26.4 KB LDS

#define M1_  0.03f
#define M2_  0.3f
#define SMP_ 0.1f

__global__ __launch_bounds__(256) void si_postprocess_kernel(
    const float* __restrict__ logits,
    const float* __restrict__ filt,
    float* __restrict__ out_pps,
    float* __restrict__ out_smp)
{
    __shared__ float tile[HN_];

    const int tid = threadIdx.x;
    const int x0  = blockIdx.x * TX_;
    const int y0  = blockIdx.y * TY_;
    const int bz  = blockIdx.z;
    const int n   = bz / (D_ / TZ_);
    const int z0  = (bz % (D_ / TZ_)) * TZ_;

    // Wave-uniform 27-tap filter -> scalar loads / SGPRs
    float w[27];
#pragma unroll
    for (int k = 0; k < 27; ++k) w[k] = filt[k];

    // ---- Phase 1: async DMA raw logits halo tile into LDS (CDNA5 path) ----
    // OOB coordinates are clamped (safe address); the value is replaced by 0
    // in phase 2, which is correct padding for both the conv and the pool.
    for (int i = tid; i < HN_; i += 256) {
        int lz = i / (HX_ * HY_);
        int r  = i - lz * (HX_ * HY_);
        int ly = r / HX_;
        int lx = r - ly * HX_;
        int gx = x0 - 1 + lx;
        int gy = y0 - 1 + ly;
        int gz = z0 - 1 + lz;
        int cx = gx < 0 ? 0 : (gx > W_ - 1 ? W_ - 1 : gx);
        int cy = gy < 0 ? 0 : (gy > H_ - 1 ? H_ - 1 : gy);
        int cz = gz < 0 ? 0 : (gz > D_ - 1 ? D_ - 1 : gz);
        long long off = (((long long)n * D_ + cz) * H_ + cy) * W_ + cx;
        unsigned long long ga = (unsigned long long)(uintptr_t)(logits + off);
        unsigned la = (unsigned)(uintptr_t)&tile[i];
        asm volatile("global_load_async_to_lds_b32 %0, %1, off"
                     :: "v"(la), "v"(ga) : "memory");
    }
#if defined(__has_builtin) && __has_builtin(__builtin_amdgcn_s_wait_asynccnt)
    __builtin_amdgcn_s_wait_asynccnt(0);
#else
    asm volatile("s_wait_asynccnt 0" ::: "memory");
#endif
    __syncthreads();

    // ---- Phase 2: sigmoid once per loaded element; zero the OOB halo ----
    for (int i = tid; i < HN_; i += 256) {
        int lz = i / (HX_ * HY_);
        int r  = i - lz * (HX_ * HY_);
        int ly = r / HX_;
        int lx = r - ly * HX_;
        bool oob = ((unsigned)(x0 - 1 + lx) >= (unsigned)W_) ||
                   ((unsigned)(y0 - 1 + ly) >= (unsigned)H_) ||
                   ((unsigned)(z0 - 1 + lz) >= (unsigned)D_);
        float v = tile[i];
        float s = 1.0f / (1.0f + __expf(-v));
        tile[i] = oob ? 0.0f : s;
    }
    __syncthreads();

    // ---- Phase 3: sliding-window stencil along z (3-plane register rotation)
    const long long obase = (((long long)n * D_ + z0) * H_ + y0) * W_ + x0;
    const int tx = tid & (TX_ - 1);   // 0..63
    const int tq = tid >> 6;          // 0..3

#pragma unroll
    for (int yy = 0; yy < 2; ++yy) {
        const int lyy = tq + yy * 4;  // output y within tile, 0..7
        float pl[3][9];
        float pm[3];

        // Preload halo planes z=0,1
#pragma unroll
        for (int zp = 0; zp < 2; ++zp) {
            const int bs = (zp * HY_ + lyy) * HX_ + tx;
            float m = 0.0f;
#pragma unroll
            for (int dy = 0; dy < 3; ++dy)
#pragma unroll
                for (int dx = 0; dx < 3; ++dx) {
                    float v = tile[bs + dy * HX_ + dx];
                    pl[zp][dy * 3 + dx] = v;
                    m = fmaxf(m, v);
                }
            pm[zp] = m;
        }

#pragma unroll
        for (int z = 0; z < TZ_; ++z) {
            const int cur = (z + 2) % 3;
            {   // load the incoming plane (only 9 new LDS loads per voxel)
                const int bs = ((z + 2) * HY_ + lyy) * HX_ + tx;
                float m = 0.0f;
#pragma unroll
                for (int dy = 0; dy < 3; ++dy)
#pragma unroll
                    for (int dx = 0; dx < 3; ++dx) {
                        float v = tile[bs + dy * HX_ + dx];
                        pl[cur][dy * 3 + dx] = v;
                        m = fmaxf(m, v);
                    }
                pm[cur] = m;
            }
            const int i0 = z % 3, i1 = (z + 1) % 3, i2 = cur;

            float conv = 0.0f;
#pragma unroll
            for (int j = 0; j < 9; ++j) conv = fmaf(w[j],      pl[i0][j], conv);
#pragma unroll
            for (int j = 0; j < 9; ++j) conv = fmaf(w[9 + j],  pl[i1][j], conv);
#pragma unroll
            for (int j = 0; j < 9; ++j) conv = fmaf(w[18 + j], pl[i2][j], conv);

            // pool = clip(max27)  (clip is monotone, commutes with max)
            float mx   = fmaxf(pm[i0], fmaxf(pm[i1], pm[i2]));
            float pool = (mx > M1_) ? mx : 0.0f;
            float p    = pl[i1][4];  // center voxel

            // p_ps = (p==pool || p>0.3) ? conv : 0  (exact mask algebra)
            bool keep = (p == pool) || (p > M2_);
            float pps = keep ? conv : 0.0f;
            float smp = (pps > SMP_) ? 1.0f : 0.0f;

            long long o = obase + ((long long)z * H_ + lyy) * W_ + tx;
            // Outputs are write-once, never re-read: non-temporal stores keep
            // L2 free for the input halo reuse (ISA 7.3 store TH = NT).
            __builtin_nontemporal_store(pps, &out_pps[o]);
            __builtin_nontemporal_store(smp, &out_smp[o]);
        }
    }
}

extern "C" void kernel_launch(void* const* d_in, const int* in_sizes, int n_in,
                              void* d_out, int out_size, void* d_ws, size_t ws_size,
                              hipStream_t stream) {
    (void)in_sizes; (void)n_in; (void)out_size; (void)d_ws; (void)ws_size;
    const float* logits = (const float*)d_in[0];
    const float* filt   = (const float*)d_in[1];
    float* out_pps = (float*)d_out;
    float* out_smp = (float*)d_out + (long long)TOT_;

    dim3 grid(W_ / TX_, H_ / TY_, (D_ / TZ_) * N_);  // 4 x 32 x 48
    si_postprocess_kernel<<<grid, 256, 0, stream>>>(logits, filt, out_pps, out_smp);
}